// MHNA_10505490006116
// MI455X (gfx1250) — compile-verified
//
#include <hip/hip_runtime.h>
#include <hip/hip_bf16.h>

// ---------------------------------------------------------------------------
// Chunked linear attention for MI455X (gfx1250, wave32)
//   fp32 WMMA (V_WMMA_F32_16X16X4_F32) + TDM tensor_load_to_lds staging
//   B=2, S=2048, D=1024, H=16, d=64, chunk C=64, NC=32 chunks
// ---------------------------------------------------------------------------

typedef float v2f __attribute__((ext_vector_type(2)));
typedef float v4f __attribute__((ext_vector_type(4)));
typedef float v8f __attribute__((ext_vector_type(8)));
typedef unsigned int u32x4 __attribute__((ext_vector_type(4)));
typedef int i32x4 __attribute__((ext_vector_type(4)));
typedef int i32x8 __attribute__((ext_vector_type(8)));

#define DM    1024      // D_MODEL / D_QK
#define BS    4096      // B*S tokens
#define HH    16        // heads
#define NC    32        // chunks per (b,h)
#define QKVLD 3072      // qkv row stride (q:0..1023, k:1024..2047, v:2048..3071)
#define LP    66        // padded LDS row stride (floats): 64 + TDM pad of 2 dwords

static __device__ __forceinline__ v8f wmma4(v2f a, v2f b, v8f c) {
  // D = A(16x4,f32) * B(4x16,f32) + C(16x16,f32)
  return __builtin_amdgcn_wmma_f32_16x16x4_f32(false, a, false, b, (short)0, c,
                                               false, false);
}

// TDM: DMA a 2D tile (tile_x contiguous elems * tile_y rows, f32) into LDS.
// pad!=0: insert 2 dwords of LDS padding every 64 dwords -> row stride 66 floats.
// Descriptor bitfields per CDNA5 ISA sec 8.3-8.5 (D# groups 0/1; groups 2/3 NULL-ish).
static __device__ __forceinline__ void tdm_load_2d(unsigned lds_off_bytes,
                                                   const float* gsrc,
                                                   unsigned tile_x, unsigned tile_y,
                                                   unsigned stride_elems, int pad)
{
  unsigned long long ga = (unsigned long long)(__UINTPTR_TYPE__)gsrc;
  u32x4 g0;
  g0[0] = 1u;                                         // count=1 valid, load, user
  g0[1] = lds_off_bytes;                              // lds_addr
  g0[2] = (unsigned)(ga & 0xffffffffull);             // global_addr[31:0]
  g0[3] = (unsigned)((ga >> 32) & 0x1ffffffull)       // global_addr[56:32]
        | (2u << 30);                                 // type = 2 (image)
  unsigned d0 = (2u << 16);                           // data_size = 4 bytes
  if (pad) d0 |= (1u << 20) | (5u << 22) | (1u << 25);// pad: every 64 dw, +2 dw
  i32x8 g1;
  g1[0] = (int)d0;
  g1[1] = (int)((stride_elems & 0xffffu) << 16);      // tensor_dim0[15:0]
  g1[2] = (int)(((stride_elems >> 16) & 0xffffu)      // tensor_dim0[31:16]
        | ((tile_y & 0xffffu) << 16));                // tensor_dim1[15:0]
  g1[3] = (int)((tile_x & 0xffffu) << 16);            // tile_dim0
  g1[4] = (int)(tile_y & 0xffffu);                    // tile_dim1 (tile_dim2=0)
  g1[5] = (int)stride_elems;                          // tensor_dim0_stride[31:0]
  g1[6] = 0;                                          // stride hi / dim1_stride lo
  g1[7] = 0;
  i32x4 z4 = {};
#if __clang_major__ >= 23
  i32x8 z8 = {};
  __builtin_amdgcn_tensor_load_to_lds(g0, g1, z4, z4, z8, 0);
#else
  __builtin_amdgcn_tensor_load_to_lds(g0, g1, z4, z4, 0);
#endif
}

static __device__ __forceinline__ unsigned lds_off(const void* p) {
  // generic shared-aperture address: low 32 bits are the LDS byte offset
  return (unsigned)(__UINTPTR_TYPE__)p;
}

// ---------------------------------------------------------------------------
// Block GEMM: 8 waves compute 128x64 of out = A @ W^T + bias.
// W strip (64 rows x K) staged through LDS by TDM, double-buffered (KT=64).
// ---------------------------------------------------------------------------
static __device__ __forceinline__ void gemm_block(
    const float* __restrict__ A, int lda,
    const float* __restrict__ W, int ldw,
    const float* __restrict__ bias,
    float* __restrict__ out, int ldo,
    int row0blk, int wrow0, int ocol0, int K,
    float* Wlds /* 2 buffers of 64*LP floats */)
{
  const int lane = threadIdx.x & 31, wave = threadIdx.x >> 5;
  const int l15 = lane & 15;
  const int kk  = 2 * (lane >> 4);
  const int row0 = row0blk + wave * 16;
  const float* arow = A + (size_t)(row0 + l15) * lda;
  const int KT = 64;
  const int nkt = K / KT;
  const unsigned lbase = lds_off(Wlds);
  const float* wstrip = W + (size_t)wrow0 * ldw;

  if (wave == 0) {
    tdm_load_2d(lbase, wstrip, KT, 64, (unsigned)ldw, 1);
    __builtin_amdgcn_s_wait_tensorcnt(0);
  }
  __syncthreads();

  v8f acc[4] = {};
  for (int kt = 0; kt < nkt; ++kt) {
    const int cur = kt & 1;
    if (wave == 0 && kt + 1 < nkt)
      tdm_load_2d(lbase + (unsigned)((cur ^ 1) * 64 * LP * 4),
                  wstrip + (kt + 1) * KT, KT, 64, (unsigned)ldw, 1);
    if (kt + 1 < nkt) {                       // keep A rows warm in WGP caches
      __builtin_prefetch(arow + (kt + 1) * KT, 0, 3);
      __builtin_prefetch(arow + (kt + 1) * KT + 32, 0, 3);
    }
    const float* wl = Wlds + cur * (64 * LP);
    const int kbase = kt * KT;
#pragma unroll
    for (int k0 = 0; k0 < KT; k0 += 4) {
      v2f a  = *(const v2f*)(arow + kbase + k0 + kk);
      v2f b0 = *(const v2f*)(wl + ( 0 + l15) * LP + k0 + kk);
      v2f b1 = *(const v2f*)(wl + (16 + l15) * LP + k0 + kk);
      v2f b2 = *(const v2f*)(wl + (32 + l15) * LP + k0 + kk);
      v2f b3 = *(const v2f*)(wl + (48 + l15) * LP + k0 + kk);
      acc[0] = wmma4(a, b0, acc[0]);
      acc[1] = wmma4(a, b1, acc[1]);
      acc[2] = wmma4(a, b2, acc[2]);
      acc[3] = wmma4(a, b3, acc[3]);
    }
    if (kt + 1 < nkt) {
      if (wave == 0) __builtin_amdgcn_s_wait_tensorcnt(0);
      __syncthreads();
    }
  }

  const int rbase = row0 + 8 * (lane >> 4);
#pragma unroll
  for (int t = 0; t < 4; ++t) {
    float bv = bias[wrow0 + 16 * t + l15];
#pragma unroll
    for (int r = 0; r < 8; ++r)
      out[(size_t)(rbase + r) * ldo + ocol0 + 16 * t + l15] = acc[t][r] + bv;
  }
}

// ---------------- Kernel 1: fused QKV projection ---------------------------
// grid = 32 m-blocks * 48 n-strips, block = 256 (8 waves)
__global__ void k_proj_qkv(const float* __restrict__ x,
                           const float* __restrict__ Wqk, const float* __restrict__ bqk,
                           const float* __restrict__ Wv,  const float* __restrict__ bv,
                           float* __restrict__ qkv)
{
  __shared__ float Wlds[2 * 64 * LP];
  const int mblk = blockIdx.x & 31;
  const int nstrip = blockIdx.x >> 5;
  const int col0 = nstrip * 64;
  const float* W; const float* bias; int wrow0;
  if (col0 < 2048) { W = Wqk; bias = bqk; wrow0 = col0; }
  else             { W = Wv;  bias = bv;  wrow0 = col0 - 2048; }
  gemm_block(x, DM, W, DM, bias, qkv, QKVLD, mblk * 128, wrow0, col0, DM, Wlds);
}

// ---------------- Kernel 2: n = exp(-(x @ Wn^T + b)) -----------------------
__global__ void k_proj_n(const float* __restrict__ x,
                         const float* __restrict__ Wn, const float* __restrict__ bn,
                         float* __restrict__ nbuf)
{
  const int idx = blockIdx.x * 256 + threadIdx.x;   // 0..65535
  const int tok = idx >> 4, h = idx & 15;
  const float* xr = x + (size_t)tok * DM;
  const float* wr = Wn + (size_t)h * DM;
  float s = bn[h];
#pragma unroll 4
  for (int k = 0; k < DM; k += 4) {
    v4f xv = *(const v4f*)(xr + k);
    v4f wv = *(const v4f*)(wr + k);
    s = fmaf(xv[0], wv[0], fmaf(xv[1], wv[1], fmaf(xv[2], wv[2], fmaf(xv[3], wv[3], s))));
  }
  nbuf[idx] = expf(-s);
}

// ---------------- Kernel 3: per-chunk KV = K^T V (64x64) -------------------
// grid = B*H*NC = 1024, block = 128 (4 waves). K/V chunk tiles staged via TDM.
__global__ void k_chunk_kv(const float* __restrict__ qkv, float* __restrict__ kv)
{
  __shared__ float Kl[64 * 64];   // rows read along columns per lane -> packed ok
  __shared__ float Vl[64 * 64];
  const int lane = threadIdx.x & 31, wave = threadIdx.x >> 5;
  const int bh = blockIdx.x >> 5, c = blockIdx.x & 31;
  const int b = bh >> 4, h = bh & 15;
  const size_t cb = (size_t)b * 2048 + (size_t)c * 64;  // token base of chunk

  if (wave == 0 && threadIdx.x == 0) { /* wave-level op; keep wave0 path */ }
  if (wave == 0) {
    const float* kp = qkv + cb * QKVLD + 1024 + h * 64;
    const float* vp = qkv + cb * QKVLD + 2048 + h * 64;
    tdm_load_2d(lds_off(Kl), kp, 64, 64, QKVLD, 0);
    tdm_load_2d(lds_off(Vl), vp, 64, 64, QKVLD, 0);
    __builtin_amdgcn_s_wait_tensorcnt(0);
  }
  __syncthreads();

  const int l15 = lane & 15;
  const int kk  = 2 * (lane >> 4);
  const int dd  = wave * 16 + l15;                      // A row (d index)

  v8f acc[4] = {};
#pragma unroll
  for (int s0 = 0; s0 < 64; s0 += 4) {
    const int sA = s0 + kk;
    v2f a = { Kl[sA * 64 + dd], Kl[(sA + 1) * 64 + dd] };   // A[d][s]
#pragma unroll
    for (int t = 0; t < 4; ++t) {
      const int tc = 16 * t + l15;
      v2f bb = { Vl[sA * 64 + tc], Vl[(sA + 1) * 64 + tc] }; // B[s][t]
      acc[t] = wmma4(a, bb, acc[t]);
    }
  }
  float* dst = kv + ((size_t)bh * NC + c) * 4096;
  const int rbase = wave * 16 + 8 * (lane >> 4);
#pragma unroll
  for (int t = 0; t < 4; ++t)
#pragma unroll
    for (int r = 0; r < 8; ++r)
      dst[(rbase + r) * 64 + 16 * t + l15] = acc[t][r];
}

// ---------------- Kernel 4: exclusive prefix scan of chunk states ----------
__global__ void k_kv_scan(float* __restrict__ kv)
{
  const int bh = blockIdx.x;
  float* base = kv + (size_t)bh * NC * 4096 + threadIdx.x * 16;
  v4f run0 = {}, run1 = {}, run2 = {}, run3 = {};
  for (int c = 0; c < NC; ++c) {
    float* p = base + (size_t)c * 4096;
    v4f t0 = *(v4f*)(p + 0),  t1 = *(v4f*)(p + 4);
    v4f t2 = *(v4f*)(p + 8),  t3 = *(v4f*)(p + 12);
    *(v4f*)(p + 0)  = run0;  *(v4f*)(p + 4)  = run1;
    *(v4f*)(p + 8)  = run2;  *(v4f*)(p + 12) = run3;
    run0 += t0; run1 += t1; run2 += t2; run3 += t3;
  }
}

// ---------------- Kernel 5: O = n * (tril(QK^T) V + Q state) ---------------
// grid = B*H*NC = 1024, block = 128 (4 waves). K/V tiles staged via TDM;
// K tile gets TDM pad (per-lane row access), V tile packed (per-lane col access).
__global__ void k_attn(const float* __restrict__ qkv,
                       const float* __restrict__ kvstate,
                       const float* __restrict__ nbuf,
                       float* __restrict__ attn)
{
  __shared__ float Kl[64 * LP];   // padded: B frags index rows per lane
  __shared__ float Vl[64 * 64];   // packed: B frags index cols per lane
  __shared__ float P[64 * LP];    // padded: A frags index rows per lane
  const int lane = threadIdx.x & 31, wave = threadIdx.x >> 5;
  const int bh = blockIdx.x >> 5, c = blockIdx.x & 31;
  const int b = bh >> 4, h = bh & 15;
  const size_t cb = (size_t)b * 2048 + (size_t)c * 64;
  const float* st = kvstate + ((size_t)bh * NC + c) * 4096;

  if (wave == 0) {
    const float* kp = qkv + cb * QKVLD + 1024 + h * 64;
    const float* vp = qkv + cb * QKVLD + 2048 + h * 64;
    tdm_load_2d(lds_off(Kl), kp, 64, 64, QKVLD, 1);
    tdm_load_2d(lds_off(Vl), vp, 64, 64, QKVLD, 0);
    __builtin_amdgcn_s_wait_tensorcnt(0);
  }
  __syncthreads();

  const int l15 = lane & 15;
  const int kk  = 2 * (lane >> 4);
  const float* qrow = qkv + (cb + wave * 16 + l15) * (size_t)QKVLD + h * 64;

  // Phase 1: S_strip = Q K^T  (K-dim = head dim 64)
  v8f acc[4] = {};
#pragma unroll
  for (int d0 = 0; d0 < 64; d0 += 4) {
    v2f a = *(const v2f*)(qrow + d0 + kk);
#pragma unroll
    for (int t = 0; t < 4; ++t) {
      v2f bb = *(const v2f*)(Kl + (16 * t + l15) * LP + d0 + kk);
      acc[t] = wmma4(a, bb, acc[t]);
    }
  }

  // Phase 2: causal mask (j <= i), stage P into LDS (D-layout -> A-layout)
#pragma unroll
  for (int t = 0; t < 4; ++t)
#pragma unroll
    for (int r = 0; r < 8; ++r) {
      const int ri = wave * 16 + r + 8 * (lane >> 4);
      const int cj = 16 * t + l15;
      P[ri * LP + cj] = (cj <= ri) ? acc[t][r] : 0.0f;
    }
  __syncthreads();

  // Phase 3a: O = P @ V  (K-dim = chunk position)
  v8f o[4] = {};
  const float* prow = P + (wave * 16 + l15) * LP;
#pragma unroll
  for (int j0 = 0; j0 < 64; j0 += 4) {
    v2f a = *(const v2f*)(prow + j0 + kk);
    const int sj = j0 + kk;
#pragma unroll
    for (int t = 0; t < 4; ++t) {
      const int tc = 16 * t + l15;
      v2f bb = { Vl[sj * 64 + tc], Vl[(sj + 1) * 64 + tc] };
      o[t] = wmma4(a, bb, o[t]);
    }
  }
  // Phase 3b: O += Q @ state  (K-dim = head dim); state packed in L2
#pragma unroll
  for (int d0 = 0; d0 < 64; d0 += 4) {
    v2f a = *(const v2f*)(qrow + d0 + kk);
    const int dr = d0 + kk;
#pragma unroll
    for (int t = 0; t < 4; ++t) {
      const int tc = 16 * t + l15;
      v2f bb = { st[dr * 64 + tc], st[(dr + 1) * 64 + tc] };
      o[t] = wmma4(a, bb, o[t]);
    }
  }

  // Phase 4: gate by n and store
#pragma unroll
  for (int r = 0; r < 8; ++r) {
    const int tok = (int)cb + wave * 16 + r + 8 * (lane >> 4);
    const float nv = nbuf[tok * 16 + h];
#pragma unroll
    for (int t = 0; t < 4; ++t)
      attn[(size_t)tok * DM + h * 64 + 16 * t + l15] = o[t][r] * nv;
  }
}

// ---------------- Kernel 6: output projection ------------------------------
// grid = 32 m-blocks * 16 n-strips, block = 256
__global__ void k_proj_out(const float* __restrict__ attn,
                           const float* __restrict__ Wo, const float* __restrict__ bo,
                           float* __restrict__ out)
{
  __shared__ float Wlds[2 * 64 * LP];
  const int mblk = blockIdx.x & 31;
  const int nstrip = blockIdx.x >> 5;
  const int col0 = nstrip * 64;
  gemm_block(attn, DM, Wo, DM, bo, out, DM, mblk * 128, col0, col0, DM, Wlds);
}

// ---------------------------------------------------------------------------
extern "C" void kernel_launch(void* const* d_in, const int* in_sizes, int n_in,
                              void* d_out, int out_size, void* d_ws, size_t ws_size,
                              hipStream_t stream)
{
  const float* x     = (const float*)d_in[0];
  const float* Wqk_w = (const float*)d_in[1];
  const float* Wqk_b = (const float*)d_in[2];
  const float* Wv_w  = (const float*)d_in[3];
  const float* Wv_b  = (const float*)d_in[4];
  const float* Wn_w  = (const float*)d_in[5];
  const float* Wn_b  = (const float*)d_in[6];
  const float* Wo_w  = (const float*)d_in[7];
  const float* Wo_b  = (const float*)d_in[8];
  float* out = (float*)d_out;

  float* ws   = (float*)d_ws;
  float* qkv  = ws;                            // 4096*3072   = 12,582,912 f
  float* kv   = qkv  + (size_t)BS * QKVLD;     // 32*32*4096  =  4,194,304 f
  float* nbuf = kv   + (size_t)32 * NC * 4096; // 4096*16     =     65,536 f
  float* attn = nbuf + (size_t)BS * HH;        // 4096*1024   =  4,194,304 f

  k_proj_qkv<<<dim3(32 * 48), dim3(256), 0, stream>>>(x, Wqk_w, Wqk_b, Wv_w, Wv_b, qkv);
  k_proj_n  <<<dim3(256),     dim3(256), 0, stream>>>(x, Wn_w, Wn_b, nbuf);
  k_chunk_kv<<<dim3(32 * NC), dim3(128), 0, stream>>>(qkv, kv);
  k_kv_scan <<<dim3(32),      dim3(256), 0, stream>>>(kv);
  k_attn    <<<dim3(32 * NC), dim3(128), 0, stream>>>(qkv, kv, nbuf, attn);
  k_proj_out<<<dim3(32 * 16), dim3(256), 0, stream>>>(attn, Wo_w, Wo_b, out);
}